// GNN_80882824118831
// MI455X (gfx1250) — compile-verified
//
#include <hip/hip_runtime.h>

typedef __attribute__((ext_vector_type(2))) float v2f;
typedef __attribute__((ext_vector_type(8))) float v8f;

#define HIDD 128
#define LDS_STRIDE 132  // 128 + 4 pad: lanes hit distinct LDS banks

// ---------------- degree / gcn_norm ----------------
__global__ void k_deg_init(float* __restrict__ deg, int n) {
  int i = blockIdx.x * blockDim.x + threadIdx.x;
  if (i < n) deg[i] = 1.0f;  // self-loop weight
}

__global__ void k_deg_accum(const int* __restrict__ col, const float* __restrict__ ew,
                            float* __restrict__ deg, int e) {
  int i = blockIdx.x * blockDim.x + threadIdx.x;
  if (i < e) unsafeAtomicAdd(&deg[col[i]], ew[i]);
}

__global__ void k_dinv(float* __restrict__ deg, int n) {
  int i = blockIdx.x * blockDim.x + threadIdx.x;
  if (i < n) {
    float d = deg[i];
    deg[i] = d > 0.0f ? rsqrtf(d) : 0.0f;  // in place: deg -> dinv
  }
}

// ---------------- WMMA fp32 GEMM: Out[nrows x NCOLS] = X[nrows x 128] @ W[128 x NCOLS] (+bias) --
// NCOLS is compile-time so all B/D addresses become constant instruction offsets
// (24-bit IOFFSET covers 125*2048*4 ~= 1 MB), removing 64-bit address VALU from the loop.
// block = 256 threads = 8 waves; block covers 16 rows x 128 cols; wave -> one 16x16 tile.
template <int NCOLS>
__global__ __launch_bounds__(256) void k_wmma_gemm(
    const float* __restrict__ X, const int* __restrict__ gather,
    const float* __restrict__ W, const float* __restrict__ bias,
    float* __restrict__ Out, int nrows) {
  __shared__ float As[16 * LDS_STRIDE];

  const int tid  = threadIdx.x;
  const int wave = tid >> 5;
  const int lane = tid & 31;
  const int lm   = lane & 15;   // N (or M) index within tile
  const int half = lane >> 4;   // lane half selects K pair / row offset
  const int row0 = blockIdx.x << 4;
  const int col0 = (blockIdx.y << 7) + (wave << 4);

  // Stage A tile: 16 rows x 128 fp32. 256 threads x 8 floats (2x b128 loads each).
  {
    const int rl   = tid >> 4;        // 0..15 local row
    const int coff = (tid & 15) << 3; // 0..120 step 8
    int r = row0 + rl;
    if (r >= nrows) r = nrows - 1;
    if (gather) r = gather[r];  // fused embedding-lookup for layer 1
    const float4* src = reinterpret_cast<const float4*>(X + (size_t)r * 128 + coff);
    float4* dst = reinterpret_cast<float4*>(&As[rl * LDS_STRIDE + coff]);
    dst[0] = src[0];
    dst[1] = src[1];
  }
  __syncthreads();

  // Per-lane bases; inner loop uses only compile-time offsets from these.
  const float* __restrict__ wp = W + (half << 1) * NCOLS + col0 + lm;  // B[K=2*half][N=lm]
  const float* __restrict__ ap = &As[lm * LDS_STRIDE + (half << 1)];   // A[M=lm][K=2*half]

  v8f acc = {};
#pragma unroll
  for (int k0 = 0; k0 < 128; k0 += 4) {
    v2f a, b;
    // A 16x4: lane lm holds M=lm; VGPR j holds K = 2*half + j  (ISA 7.12.2)
    a[0] = ap[k0];
    a[1] = ap[k0 + 1];
    // B 4x16: lane lm holds N=lm; VGPR j holds K = 2*half + j
    b[0] = wp[k0 * NCOLS];
    b[1] = wp[k0 * NCOLS + NCOLS];
    acc = __builtin_amdgcn_wmma_f32_16x16x4_f32(false, a, false, b,
                                                (short)0, acc, false, false);
  }

  const float bc = bias ? bias[col0 + lm] : 0.0f;
  float* __restrict__ op = Out + (size_t)(row0 + (half << 3)) * NCOLS + col0 + lm;
#pragma unroll
  for (int rr = 0; rr < 8; ++rr) {
    // D: VGPR rr -> row rr + 8*half, col lm
    if (row0 + (half << 3) + rr < nrows)
      op[rr * NCOLS] = acc[rr] + bc;
  }
}

// ---------------- aggregation ----------------
// out[i] = h[i] * dinv[i]^2   (self loop, ew=1) -- also initializes the accumulator
__global__ void k_selfloop_init(const float* __restrict__ h, const float* __restrict__ dinv,
                                float* __restrict__ out, int n) {
  int idx = blockIdx.x * blockDim.x + threadIdx.x;  // n * 32 threads (float4 each)
  if (idx >= n * 32) return;
  const int node = idx >> 5;
  const int c    = idx & 31;
  float s = dinv[node];
  s = s * s;
  float4 v = reinterpret_cast<const float4*>(h)[(size_t)node * 32 + c];
  v.x *= s; v.y *= s; v.z *= s; v.w *= s;
  reinterpret_cast<float4*>(out)[(size_t)node * 32 + c] = v;
}

// out[col[e]] += h[row[e]] * (dinv[row]*ew*dinv[col]); one wave per edge, float4 per lane.
// h and out are 51 MB each -> L2-resident; atomics resolve in L2.
__global__ __launch_bounds__(256) void k_scatter(
    const float* __restrict__ h, const int* __restrict__ row, const int* __restrict__ col,
    const float* __restrict__ ew, const float* __restrict__ dinv,
    float* __restrict__ out, int e) {
  const int eid = (blockIdx.x << 3) + (threadIdx.x >> 5);
  if (eid >= e) return;
  const int lane = threadIdx.x & 31;
  const int r = row[eid];
  const int c = col[eid];
  const float nrm = dinv[r] * ew[eid] * dinv[c];
  float4 v = reinterpret_cast<const float4*>(h)[(size_t)r * 32 + lane];
  float* dst = out + (size_t)c * 128 + (lane << 2);
  unsafeAtomicAdd(dst + 0, v.x * nrm);
  unsafeAtomicAdd(dst + 1, v.y * nrm);
  unsafeAtomicAdd(dst + 2, v.z * nrm);
  unsafeAtomicAdd(dst + 3, v.w * nrm);
}

__global__ void k_bias_relu(float* __restrict__ x, const float* __restrict__ b, int total) {
  int i = blockIdx.x * blockDim.x + threadIdx.x;
  if (i < total) {
    float v = x[i] + b[i & 127];
    x[i] = v > 0.0f ? v : 0.0f;
  }
}

extern "C" void kernel_launch(void* const* d_in, const int* in_sizes, int n_in,
                              void* d_out, int out_size, void* d_ws, size_t ws_size,
                              hipStream_t stream) {
  const int*   node_idx = (const int*)d_in[0];
  const int*   eidx     = (const int*)d_in[1];   // [2, E]
  const float* ew       = (const float*)d_in[2];
  const float* emb      = (const float*)d_in[3];
  const float* W1       = (const float*)d_in[4];
  const float* b1       = (const float*)d_in[5];
  const float* W2       = (const float*)d_in[6];
  const float* b2       = (const float*)d_in[7];
  const float* Wout     = (const float*)d_in[8];
  const float* bout     = (const float*)d_in[9];
  float*       out      = (float*)d_out;

  const int n = in_sizes[0];
  const int e = in_sizes[2];
  const int* row = eidx;
  const int* col = eidx + e;

  // workspace: dinv[n] | buf0[n*128] | buf1[n*128]
  float* dinv = (float*)d_ws;
  float* buf0 = dinv + (((size_t)n + 255) & ~(size_t)255);
  float* buf1 = buf0 + (size_t)n * 128;

  const int T = 256;
  const int feat_blocks = (int)(((size_t)n * 128 + T - 1) / T);
  const int f4_blocks   = (n * 32 + T - 1) / T;

  // gcn_norm (shared by both layers)
  k_deg_init <<<(n + T - 1) / T, T, 0, stream>>>(dinv, n);
  k_deg_accum<<<(e + T - 1) / T, T, 0, stream>>>(col, ew, dinv, e);
  k_dinv     <<<(n + T - 1) / T, T, 0, stream>>>(dinv, n);

  dim3 g_hid((n + 15) / 16, HIDD / 128);

  // layer 1: h = emb[node_idx] @ W1 ; aggregate ; +b1 ; relu
  k_wmma_gemm<HIDD><<<g_hid, T, 0, stream>>>(emb, node_idx, W1, nullptr, buf0, n);
  k_selfloop_init<<<f4_blocks, T, 0, stream>>>(buf0, dinv, buf1, n);
  k_scatter      <<<(e + 7) / 8, T, 0, stream>>>(buf0, row, col, ew, dinv, buf1, e);
  k_bias_relu    <<<feat_blocks, T, 0, stream>>>(buf1, b1, n * 128);

  // layer 2: h2 = x1 @ W2 ; aggregate ; +b2 ; relu   (buffers ping-pong)
  k_wmma_gemm<HIDD><<<g_hid, T, 0, stream>>>(buf1, nullptr, W2, nullptr, buf0, n);
  k_selfloop_init<<<f4_blocks, T, 0, stream>>>(buf0, dinv, buf1, n);
  k_scatter      <<<(e + 7) / 8, T, 0, stream>>>(buf0, row, col, ew, dinv, buf1, e);
  k_bias_relu    <<<feat_blocks, T, 0, stream>>>(buf1, b2, n * 128);

  // output: out = x2 @ Wout + bout   [n, 2048] -- 819 MB store, the HBM-bound stage
  dim3 g_out((n + 15) / 16, 2048 / 128);
  k_wmma_gemm<2048><<<g_out, T, 0, stream>>>(buf1, nullptr, Wout, bout, out, n);
}